// Point_Transformer_partseg_33552284516957
// MI455X (gfx1250) — compile-verified
//
#include <hip/hip_runtime.h>
#include <hip/hip_bf16.h>

typedef float v2f __attribute__((ext_vector_type(2)));
typedef float v8f __attribute__((ext_vector_type(8)));

#define NPTS 2048
#define BNEPS 1e-5f

// ---------------------------------------------------------------------------
// Generic 1x1-conv GEMM:  Y[b,o,n] = act(bn(sum_c W[o,c]*X[b,c,n] + bias))
// One wave32 computes a 16(o) x 16(n) tile with V_WMMA_F32_16X16X4_F32.
// Fragment layouts per CDNA5 ISA 7.12.2:
//   A (16x4): lanes 0-15 hold M=0..15 with K={k0,k0+1}; lanes 16-31 same M, K={k0+2,k0+3}
//   B (4x16): lanes 0-15 hold N=0..15 with K={k0,k0+1}; lanes 16-31 same N, K={k0+2,k0+3}
//   C/D (16x16): VGPR r -> M=r (lanes 0-15) / M=8+r (lanes 16-31), N=lane&15
// ---------------------------------------------------------------------------
__global__ __launch_bounds__(128) void gemm_bn_act(
    const float* __restrict__ W, int ldw, int K,
    const float* __restrict__ X, long xb,
    float* __restrict__ Y, long yb, int O,
    const float* __restrict__ bias,    // [O] or null
    const float* __restrict__ pbias,   // [B,O] or null (per-batch bias)
    const float* __restrict__ bn,      // [4,O] stacked g,b,m,v or null
    int act)                           // 0 none, 1 relu, 2 leaky(0.2)
{
    const int lane  = threadIdx.x & 31;
    const int wave  = threadIdx.x >> 5;             // 4 waves: 16-wide n sub-tiles
    const int b     = blockIdx.z;
    const int oBase = blockIdx.y * 16;
    const int nBase = blockIdx.x * 64 + wave * 16;
    const int row   = lane & 15;
    const int hi    = lane >> 4;                    // K half select

    const float* Wr = W + (size_t)(oBase + row) * ldw;
    const float* Xc = X + (size_t)b * xb + nBase + row;   // column index == lane&15

    v8f acc = {0.f, 0.f, 0.f, 0.f, 0.f, 0.f, 0.f, 0.f};
    for (int k0 = 0; k0 < K; k0 += 4) {
        const int ka = k0 + hi * 2;
        v2f af, bf;
        af.x = Wr[ka];
        af.y = Wr[ka + 1];
        bf.x = Xc[(size_t)ka * NPTS];
        bf.y = Xc[(size_t)(ka + 1) * NPTS];
        acc = __builtin_amdgcn_wmma_f32_16x16x4_f32(false, af, false, bf,
                                                    (short)0, acc, false, false);
    }

    float* Yp = Y + (size_t)b * yb + nBase + row;
#pragma unroll
    for (int r = 0; r < 8; ++r) {
        const int o = oBase + hi * 8 + r;
        float v = acc[r];
        if (bias)  v += bias[o];
        if (pbias) v += pbias[b * O + o];
        if (bn) {
            const float g = bn[o], be = bn[O + o], mu = bn[2 * O + o], va = bn[3 * O + o];
            v = (v - mu) * __frsqrt_rn(va + BNEPS) * g + be;
        }
        if (act == 1)      v = fmaxf(v, 0.f);
        else if (act == 2) v = (v > 0.f) ? v : 0.2f * v;
        Yp[(size_t)o * NPTS] = v;
    }
}

// ---------------------------------------------------------------------------
// SA helpers.  q is tied to k (same weight) => logits L[n,m] = sum_j k[j,n]k[j,m]
// k: [B,4,2048] scratch.  Attention matrix is never materialized.
// ---------------------------------------------------------------------------
__global__ __launch_bounds__(256) void sa_k(const float* __restrict__ hin, long hb,
                                            const float* __restrict__ qkw,
                                            float* __restrict__ kbuf)
{
    const int id = blockIdx.x * 256 + threadIdx.x;   // 16*4*2048 total
    const int b = id >> 13, j = (id >> 11) & 3, n = id & 2047;
    const float* hp = hin + (size_t)b * hb + n;
    const float* wr = qkw + j * 16;
    float s = 0.f;
#pragma unroll
    for (int c = 0; c < 16; ++c) s += wr[c] * hp[(size_t)c * NPTS];
    kbuf[((size_t)b * 4 + j) * NPTS + n] = s;
}

// Row softmax stats: online max/sum over m for each row n.
__global__ __launch_bounds__(256) void sa_rows(const float* __restrict__ kbuf,
                                               float* __restrict__ rmax,
                                               float* __restrict__ rsinv)
{
    __shared__ float sk[4][NPTS];
    const int b = blockIdx.y;
    const float* kb = kbuf + (size_t)b * 4 * NPTS;
    for (int i = threadIdx.x; i < 4 * NPTS; i += 256) sk[i >> 11][i & 2047] = kb[i];
    __syncthreads();
    const int n = blockIdx.x * 256 + threadIdx.x;
    const float k0 = sk[0][n], k1 = sk[1][n], k2 = sk[2][n], k3 = sk[3][n];
    float mx = -INFINITY, sum = 0.f;
    for (int m = 0; m < NPTS; ++m) {
        const float l = k0 * sk[0][m] + k1 * sk[1][m] + k2 * sk[2][m] + k3 * sk[3][m];
        if (l > mx) { sum = sum * __expf(mx - l) + 1.f; mx = l; }
        else        { sum += __expf(l - mx); }
    }
    rmax [b * NPTS + n] = mx;
    rsinv[b * NPTS + n] = 1.f / sum;
}

// Column sums of the row-softmaxed matrix -> 1/(1e-9 + colsum)
__global__ __launch_bounds__(256) void sa_cols(const float* __restrict__ kbuf,
                                               const float* __restrict__ rmax,
                                               const float* __restrict__ rsinv,
                                               float* __restrict__ icol)
{
    __shared__ float sk[4][NPTS];
    __shared__ float sm[NPTS], si[NPTS];
    const int b = blockIdx.y;
    const float* kb = kbuf + (size_t)b * 4 * NPTS;
    for (int i = threadIdx.x; i < 4 * NPTS; i += 256) sk[i >> 11][i & 2047] = kb[i];
    for (int i = threadIdx.x; i < NPTS; i += 256) {
        sm[i] = rmax[b * NPTS + i];
        si[i] = rsinv[b * NPTS + i];
    }
    __syncthreads();
    const int m = blockIdx.x * 256 + threadIdx.x;
    const float k0 = sk[0][m], k1 = sk[1][m], k2 = sk[2][m], k3 = sk[3][m];
    float s = 0.f;
    for (int n = 0; n < NPTS; ++n) {
        const float l = sk[0][n] * k0 + sk[1][n] * k1 + sk[2][n] * k2 + sk[3][n] * k3;
        s += __expf(l - sm[n]) * si[n];
    }
    icol[b * NPTS + m] = 1.f / (1e-9f + s);
}

// ---------------------------------------------------------------------------
// x_r = V(16xN) * P(Nx16-col-tile) via WMMA: each wave owns 16 columns, the
// probability B-fragment is computed in-register (2 exps/lane per K-step of 4),
// A-fragment is a b64 LDS read of the staged V tile.  After the K loop the
// x_r tile is transposed through LDS (region shared with the V tile) so the
// fused trans-conv + BN + ReLU + residual epilogue sees all 16 channels.
// Dynamic LDS: sk 4*2048 | sm 2048 | si 2048 | scratch 2176 floats = 57856 B.
// ---------------------------------------------------------------------------
#define SA_SMEM_FLOATS (4 * NPTS + NPTS + NPTS + 2176)
__global__ __launch_bounds__(256) void sa_out_wmma(
    const float* __restrict__ kbuf, const float* __restrict__ rmax,
    const float* __restrict__ rsinv, const float* __restrict__ icol,
    const float* __restrict__ vbuf,
    const float* __restrict__ hin, long hb,
    const float* __restrict__ tw, const float* __restrict__ tb,
    const float* __restrict__ bnp,
    float* __restrict__ hout, long ob)
{
    extern __shared__ float smem[];
    float (*sk)[NPTS] = (float (*)[NPTS])smem;     // [4][2048]
    float* sm      = smem + 4 * NPTS;              // [2048]
    float* si      = sm + NPTS;                    // [2048]
    float* scratch = si + NPTS;                    // vt [16][130] / xrt [8][16][17]

    const int tid  = threadIdx.x;
    const int lane = tid & 31;
    const int wave = tid >> 5;                     // 8 waves -> 128 columns/block
    const int row  = lane & 15;
    const int hi   = lane >> 4;
    const int b    = blockIdx.y;
    const int mW   = blockIdx.x * 128 + wave * 16; // wave column base
    const int m    = mW + row;                     // this lane's column (B frag N)

    const float* kb = kbuf + (size_t)b * 4 * NPTS;
    for (int i = tid; i < 4 * NPTS; i += 256) sk[i >> 11][i & 2047] = kb[i];
    for (int i = tid; i < NPTS; i += 256) {
        sm[i] = rmax[b * NPTS + i];
        si[i] = rsinv[b * NPTS + i];
    }
    __syncthreads();

    const float km0 = sk[0][m], km1 = sk[1][m], km2 = sk[2][m], km3 = sk[3][m];
    const float* vb = vbuf + (size_t)b * 16 * NPTS;

    v8f acc = {0.f, 0.f, 0.f, 0.f, 0.f, 0.f, 0.f, 0.f};
    for (int nt = 0; nt < NPTS; nt += 128) {
        __syncthreads();
        for (int i = tid; i < 16 * 128; i += 256)           // stage V tile [16][130]
            scratch[(i >> 7) * 130 + (i & 127)] = vb[(size_t)(i >> 7) * NPTS + nt + (i & 127)];
        __syncthreads();
#pragma unroll 4
        for (int k0 = 0; k0 < 128; k0 += 4) {
            const int ja = k0 + hi * 2;                     // local K pair
            const int ka = nt + ja;                         // global n
            v2f af, bf;
            af.x = scratch[row * 130 + ja];
            af.y = scratch[row * 130 + ja + 1];
            const float l0 = km0 * sk[0][ka]     + km1 * sk[1][ka]
                           + km2 * sk[2][ka]     + km3 * sk[3][ka];
            const float l1 = km0 * sk[0][ka + 1] + km1 * sk[1][ka + 1]
                           + km2 * sk[2][ka + 1] + km3 * sk[3][ka + 1];
            bf.x = __expf(l0 - sm[ka])     * si[ka];
            bf.y = __expf(l1 - sm[ka + 1]) * si[ka + 1];
            acc = __builtin_amdgcn_wmma_f32_16x16x4_f32(false, af, false, bf,
                                                        (short)0, acc, false, false);
        }
    }

    // transpose x_r tile through LDS (includes column renorm 1/(1e-9+colsum))
    const float ic = icol[b * NPTS + m];
    __syncthreads();
#pragma unroll
    for (int r = 0; r < 8; ++r) {
        const int c = hi * 8 + r;
        scratch[wave * 272 + c * 17 + row] = acc[r] * ic;
    }
    __syncthreads();

    if (tid < 128) {
        const int w = tid >> 4, cidx = tid & 15;
        const int mg = blockIdx.x * 128 + tid;              // global column
        float hv[16], u[16];
        const float* hp = hin + (size_t)b * hb + mg;
#pragma unroll
        for (int c = 0; c < 16; ++c) {
            hv[c] = hp[(size_t)c * NPTS];
            u[c]  = hv[c] - scratch[w * 272 + c * 17 + cidx];
        }
        float* op = hout + (size_t)b * ob + mg;
#pragma unroll
        for (int o = 0; o < 16; ++o) {
            float t = tb[o];
#pragma unroll
            for (int c = 0; c < 16; ++c) t += tw[o * 16 + c] * u[c];
            const float g = bnp[o], be = bnp[16 + o], mu = bnp[32 + o], va = bnp[48 + o];
            float y = (t - mu) * __frsqrt_rn(va + BNEPS) * g + be;
            y = fmaxf(y, 0.f);
            op[(size_t)o * NPTS] = hv[o] + y;
        }
    }
}

// ---------------------------------------------------------------------------
// argmax (as float) and mean over n for xqqw[b,o,:]
// ---------------------------------------------------------------------------
__global__ __launch_bounds__(256) void reduce_maxavg(const float* __restrict__ xqqw,
                                                     float* __restrict__ amaxf,
                                                     float* __restrict__ meanb)
{
    __shared__ float smax[256]; __shared__ int sidx[256]; __shared__ float ssum[256];
    const int bo = blockIdx.x;                    // b*128+o
    const float* rowp = xqqw + (size_t)bo * NPTS;
    const int tid = threadIdx.x;
    float mx = -INFINITY; int mi = 0; float sum = 0.f;
    for (int n = tid; n < NPTS; n += 256) {
        const float v = rowp[n];
        sum += v;
        if (v > mx) { mx = v; mi = n; }
    }
    smax[tid] = mx; sidx[tid] = mi; ssum[tid] = sum;
    __syncthreads();
    for (int s = 128; s > 0; s >>= 1) {
        if (tid < s) {
            ssum[tid] += ssum[tid + s];
            const float om = smax[tid + s]; const int oi = sidx[tid + s];
            if (om > smax[tid] || (om == smax[tid] && oi < sidx[tid])) {
                smax[tid] = om; sidx[tid] = oi;
            }
        }
        __syncthreads();
    }
    if (tid == 0) {
        amaxf[bo] = (float)sidx[0];
        meanb[bo] = ssum[0] * (1.0f / NPTS);
    }
}

// convs1's broadcast channels are constant over n -> fold into per-(b,o) bias.
__global__ __launch_bounds__(512) void convs1_bias(const float* __restrict__ w,    // [512,264]
                                                   const float* __restrict__ cb,   // [512]
                                                   const float* __restrict__ amaxf,// [16,128]
                                                   const float* __restrict__ meanb,// [16,128]
                                                   float* __restrict__ bias1)      // [16,512]
{
    const int b = blockIdx.x, o = threadIdx.x;
    const float* wr = w + (size_t)o * 264;
    float s = cb[o];
    for (int j = 0; j < 128; ++j) s += wr[128 + j] * amaxf[b * 128 + j];
#pragma unroll
    for (int j = 0; j < 8; ++j)  s += wr[256 + j] * meanb[128 + b * 8 + j];  // batch-1 means
    bias1[b * 512 + o] = s;
}

// ---------------------------------------------------------------------------
extern "C" void kernel_launch(void* const* d_in, const int* in_sizes, int n_in,
                              void* d_out, int out_size, void* d_ws, size_t ws_size,
                              hipStream_t stream)
{
    (void)in_sizes; (void)n_in; (void)out_size; (void)ws_size;
    const float* x        = (const float*)d_in[0];
    const float* conv1_w  = (const float*)d_in[1];
    const float* bn1      = (const float*)d_in[2];
    const float* conv2_w  = (const float*)d_in[3];
    const float* bn2      = (const float*)d_in[4];
    const float* sa_qk_w  = (const float*)d_in[5];
    const float* sa_v_w   = (const float*)d_in[6];
    const float* sa_v_b   = (const float*)d_in[7];
    const float* sa_t_w   = (const float*)d_in[8];
    const float* sa_t_b   = (const float*)d_in[9];
    const float* sa_bn    = (const float*)d_in[10];
    const float* fuse_w   = (const float*)d_in[11];
    const float* fuse_bn  = (const float*)d_in[12];
    const float* convs1_w = (const float*)d_in[13];
    const float* convs1_b = (const float*)d_in[14];
    const float* bns1     = (const float*)d_in[15];
    const float* convs2_w = (const float*)d_in[16];
    const float* convs2_b = (const float*)d_in[17];
    const float* bns2     = (const float*)d_in[18];
    const float* convs3_w = (const float*)d_in[19];
    const float* convs3_b = (const float*)d_in[20];
    float* out = (float*)d_out;

    const long B = 16, N = NPTS;
    float* ws    = (float*)d_ws;
    float* g1    = ws;                    // [16,512,2048]
    float* g2    = g1 + 16L * 512 * N;    // [16,256,2048]
    float* h1    = g2;                    // alias: dead before g2 is written
    float* xqqw  = g2;                    // alias: dead before g2 is written
    float* xcat  = g2 + 16L * 256 * N;    // [16,64,2048] concat of the 4 SA feats
    float* h2    = xcat + 16L * 64 * N;   // [16,16,2048]
    float* vbuf  = h2 + 16L * 16 * N;     // [16,16,2048]
    float* kbuf  = vbuf + 16L * 16 * N;   // [16,4,2048]
    float* rmax  = kbuf + 16L * 4 * N;    // [16,2048]
    float* rsinv = rmax + 16L * N;
    float* icol  = rsinv + 16L * N;
    float* amaxf = icol + 16L * N;        // [16,128]
    float* meanb = amaxf + 16L * 128;     // [16,128]
    float* bias1 = meanb + 16L * 128;     // [16,512]

    const dim3 gblk(128);
    auto gemm = [&](const float* W, int ldw, int K, const float* X, long xb,
                    float* Y, long yb, int O, const float* bias, const float* pbias,
                    const float* bn, int act) {
        dim3 grid((unsigned)(N / 64), (unsigned)(O / 16), (unsigned)B);
        hipLaunchKernelGGL(gemm_bn_act, grid, gblk, 0, stream,
                           W, ldw, K, X, xb, Y, yb, O, bias, pbias, bn, act);
    };

    // conv1 + bn1 + relu : [16,128,2048]
    gemm(conv1_w, 16, 16, x, 16 * N, h1, 128 * N, 128, nullptr, nullptr, bn1, 1);
    // conv2 + bn2 + relu : [16,16,2048]
    gemm(conv2_w, 128, 128, h1, 128 * N, h2, 16 * N, 16, nullptr, nullptr, bn2, 1);

    // 4 offset self-attention blocks; outputs chained through xcat channel slices
    for (int i = 0; i < 4; ++i) {
        const float* hin = (i == 0) ? h2 : (xcat + (size_t)(i - 1) * 16 * N);
        const long hb    = (i == 0) ? 16 * N : 64 * N;
        float* hout = xcat + (size_t)i * 16 * N;
        // v = v_w @ h + v_b
        gemm(sa_v_w + i * 256, 16, 16, hin, hb, vbuf, 16 * N, 16,
             sa_v_b + i * 16, nullptr, nullptr, 0);
        hipLaunchKernelGGL(sa_k, dim3(16 * 4 * NPTS / 256), dim3(256), 0, stream,
                           hin, hb, sa_qk_w + i * 64, kbuf);
        hipLaunchKernelGGL(sa_rows, dim3(8, 16), dim3(256), 0, stream, kbuf, rmax, rsinv);
        hipLaunchKernelGGL(sa_cols, dim3(8, 16), dim3(256), 0, stream, kbuf, rmax, rsinv, icol);
        hipLaunchKernelGGL(sa_out_wmma, dim3(16, 16), dim3(256),
                           SA_SMEM_FLOATS * sizeof(float), stream,
                           kbuf, rmax, rsinv, icol, vbuf, hin, hb,
                           sa_t_w + i * 256, sa_t_b + i * 16, sa_bn + i * 64,
                           hout, 64 * N);
    }

    // fuse + bn + leaky(0.2) : [16,128,2048]
    gemm(fuse_w, 64, 64, xcat, 64 * N, xqqw, 128 * N, 128, nullptr, nullptr, fuse_bn, 2);

    // broadcast channels folded into per-(b,o) bias for convs1
    hipLaunchKernelGGL(reduce_maxavg, dim3(16 * 128), dim3(256), 0, stream, xqqw, amaxf, meanb);
    hipLaunchKernelGGL(convs1_bias, dim3(16), dim3(512), 0, stream,
                       convs1_w, convs1_b, amaxf, meanb, bias1);

    // convs1 (first 128 input channels, ld 264) + folded bias + bn + relu
    gemm(convs1_w, 264, 128, xqqw, 128 * N, g1, 512 * N, 512, nullptr, bias1, bns1, 1);
    // convs2 + bias + bn + relu
    gemm(convs2_w, 512, 512, g1, 512 * N, g2, 256 * N, 256, convs2_b, nullptr, bns2, 1);
    // convs3 + bias (no bn, no act) -> output
    gemm(convs3_w, 256, 256, g2, 256 * N, out, 16 * N, 16, convs3_b, nullptr, nullptr, 0);
}